// LG_GCNConv_15238543966834
// MI455X (gfx1250) — compile-verified
//
#include <hip/hip_runtime.h>
#include <math.h>

// ---------------------------------------------------------------------------
// LG-GCNConv pipeline for MI455X (gfx1250, wave32, WMMA f32 16x16x4)
// ---------------------------------------------------------------------------

typedef float v2f __attribute__((ext_vector_type(2)));
typedef float v8f __attribute__((ext_vector_type(8)));

#define D_DIM 128
#define NB    4096   // batch graphs (reference constant B)

static __device__ __forceinline__ unsigned ordf(float f) {
  unsigned b = __float_as_uint(f);
  return (b & 0x80000000u) ? ~b : (b | 0x80000000u);
}
static __device__ __forceinline__ float unordf(unsigned u) {
  return (u & 0x80000000u) ? __uint_as_float(u ^ 0x80000000u)
                           : __uint_as_float(~u);
}
static __device__ __forceinline__ float wave_sum(float v) {
#pragma unroll
  for (int m = 16; m >= 1; m >>= 1) v += __shfl_xor(v, m, 32);
  return v;
}

// ---------------------------------------------------------------------------
// WMMA GEMM: Y[M,128] = X[M,128] @ W[128,128]  (f32, V_WMMA_F32_16X16X4_F32)
// EPI==0: plain store.  EPI==1: Y = tanh(acc + bias[col]).
// Block: 256 threads = 8 waves; wave w owns output columns [16w,16w+16).
// Each block processes a 64-row stripe; A-tile staged in LDS (stride 132 to
// avoid bank conflicts); B fragments (all 32 k-steps) held in VGPRs.
// ---------------------------------------------------------------------------
#define TILE_ROWS 64
#define LDS_STRIDE 132

template <int EPI>
__global__ __launch_bounds__(256) void gemm128_wmma(
    const float* __restrict__ X, const float* __restrict__ W,
    const float* __restrict__ bias, float* __restrict__ Y, int M) {
  __shared__ float As[TILE_ROWS * LDS_STRIDE];

  const int tid     = threadIdx.x;
  const int lane    = tid & 31;
  const int wave    = tid >> 5;      // 0..7 -> column tile
  const int h       = lane >> 4;     // half-wave (K offset selector)
  const int ln      = lane & 15;     // M / N index within tile
  const int colbase = wave * 16;

  // Preload this wave's B fragments: B(4x16) per k-step, K = 4*ks + 2h + {0,1}
  v2f bf[32];
#pragma unroll
  for (int ks = 0; ks < 32; ++ks) {
    int k = ks * 4 + 2 * h;
    bf[ks].x = W[(size_t)k * D_DIM + colbase + ln];
    bf[ks].y = W[(size_t)(k + 1) * D_DIM + colbase + ln];
  }

  const int rowblock = blockIdx.x * TILE_ROWS;

  // Cooperative load of the 64x128 A tile (zero-padded past M)
  for (int i = tid; i < TILE_ROWS * 32; i += 256) {
    int r  = i >> 5;   // row in tile
    int c4 = i & 31;   // float4 column
    int gr = rowblock + r;
    float4 v = make_float4(0.f, 0.f, 0.f, 0.f);
    if (gr < M) v = ((const float4*)(X + (size_t)gr * D_DIM))[c4];
    float* dst = &As[r * LDS_STRIDE + c4 * 4];
    dst[0] = v.x; dst[1] = v.y; dst[2] = v.z; dst[3] = v.w;
  }
  __syncthreads();

#pragma unroll
  for (int rt = 0; rt < TILE_ROWS / 16; ++rt) {
    v8f acc = {};
    const float* arow = &As[(rt * 16 + ln) * LDS_STRIDE];
#pragma unroll
    for (int ks = 0; ks < 32; ++ks) {
      int k = ks * 4 + 2 * h;
      v2f a;
      a.x = arow[k];
      a.y = arow[k + 1];
      acc = __builtin_amdgcn_wmma_f32_16x16x4_f32(
          false, a, false, bf[ks], (short)0, acc, false, false);
    }
#pragma unroll
    for (int j = 0; j < 8; ++j) {
      int gr = rowblock + rt * 16 + j + 8 * h;   // C/D layout: M = j + 8h
      if (gr < M) {
        float v = acc[j];
        if (EPI == 1) v = tanhf(v + bias[colbase + ln]);
        Y[(size_t)gr * D_DIM + colbase + ln] = v;
      }
    }
  }
}

// ---------------------------------------------------------------------------
// Small utility / graph kernels
// ---------------------------------------------------------------------------
__global__ void zero_u32(unsigned* p, size_t n) {
  size_t i = (size_t)blockIdx.x * blockDim.x + threadIdx.x;
  size_t stride = (size_t)gridDim.x * blockDim.x;
  for (; i < n; i += stride) p[i] = 0u;
}

__global__ void count_deg(const int* __restrict__ col, unsigned* __restrict__ deg,
                          int ne) {
  int i = blockIdx.x * blockDim.x + threadIdx.x;
  if (i < ne) atomicAdd(&deg[col[i]], 1u);
}

__global__ void make_dinv(const unsigned* __restrict__ deg,
                          float* __restrict__ dinv, int n) {
  int i = blockIdx.x * blockDim.x + threadIdx.x;
  if (i < n) dinv[i] = rsqrtf((float)deg[i] + 1.0f);  // +1: self-loop
}

// y = xw * dinv[node] (in place); agg = y * dinv[node] (self-loop term)
__global__ void scale_self(float* __restrict__ xw, float* __restrict__ agg,
                           const float* __restrict__ dinv, size_t n) {
  size_t i = (size_t)blockIdx.x * blockDim.x + threadIdx.x;
  if (i >= n) return;
  float d = dinv[i >> 7];
  float y = xw[i] * d;
  xw[i]  = y;
  agg[i] = y * d;
}

// wave per edge: agg[col] += y[row] * dinv[col]   (128-wide, float4/lane)
__global__ __launch_bounds__(256) void scatter_edges(
    const float* __restrict__ y, float* __restrict__ agg,
    const int* __restrict__ row, const int* __restrict__ col,
    const float* __restrict__ dinv, int ne) {
  int lane = threadIdx.x & 31;
  int e = (blockIdx.x * blockDim.x + threadIdx.x) >> 5;
  if (e >= ne) return;
  int r = row[e], c = col[e];
  // warm L2 for a row a few edges ahead (global_prefetch_b8)
  if (e + 8 < ne) __builtin_prefetch(y + (size_t)row[e + 8] * D_DIM + lane * 4, 0, 1);
  float dc = dinv[c];
  float4 v = ((const float4*)(y + (size_t)r * D_DIM))[lane];
  float* dst = agg + (size_t)c * D_DIM + lane * 4;
  atomicAdd(dst + 0, v.x * dc);
  atomicAdd(dst + 1, v.y * dc);
  atomicAdd(dst + 2, v.z * dc);
  atomicAdd(dst + 3, v.w * dc);
}

// x_out = relu(edge_attr + agg + b[col])
__global__ void bias_relu(const float* __restrict__ ea,
                          const float* __restrict__ agg,
                          const float* __restrict__ b,
                          float* __restrict__ xo, size_t n) {
  size_t i = (size_t)blockIdx.x * blockDim.x + threadIdx.x;
  if (i >= n) return;
  float v = ea[i] + agg[i] + b[i & (D_DIM - 1)];
  xo[i] = v > 0.f ? v : 0.f;
}

// wave per node: s_node = x . W_rel ; sraw = x . W_root + b_rel
__global__ __launch_bounds__(256) void node_scores(
    const float* __restrict__ x, const float* __restrict__ Wrel,
    const float* __restrict__ Wroot, const float* __restrict__ brel,
    float* __restrict__ s_node, float* __restrict__ sraw, int n) {
  int lane = threadIdx.x & 31;
  int i = (blockIdx.x * blockDim.x + threadIdx.x) >> 5;
  if (i >= n) return;
  float4 xv = ((const float4*)(x + (size_t)i * D_DIM))[lane];
  float4 wr = ((const float4*)Wrel)[lane];
  float4 wo = ((const float4*)Wroot)[lane];
  float srel  = xv.x * wr.x + xv.y * wr.y + xv.z * wr.z + xv.w * wr.w;
  float sroot = xv.x * wo.x + xv.y * wo.y + xv.z * wo.z + xv.w * wo.w;
  srel  = wave_sum(srel);
  sroot = wave_sum(sroot);
  if (lane == 0) {
    s_node[i] = srel;
    sraw[i]   = sroot + brel[0];
  }
}

__global__ void scatter_scalar(const float* __restrict__ s_node,
                               float* __restrict__ sraw,
                               const int* __restrict__ row,
                               const int* __restrict__ col, int ne) {
  int e = blockIdx.x * blockDim.x + threadIdx.x;
  if (e < ne) atomicAdd(&sraw[col[e]], s_node[row[e]]);
}

// zero gx_t (NB*D), reset smax/z
__global__ void init_seg(float* __restrict__ gx, unsigned* __restrict__ smax,
                         float* __restrict__ z, int nbd) {
  int i = blockIdx.x * blockDim.x + threadIdx.x;
  if (i < nbd) gx[i] = 0.f;
  if (i < NB) { smax[i] = 0u; z[i] = 0.f; }
}

__global__ void seg_max(const float* __restrict__ sraw,
                        const int* __restrict__ batch,
                        unsigned* __restrict__ smax, int n) {
  int i = blockIdx.x * blockDim.x + threadIdx.x;
  if (i < n) atomicMax(&smax[batch[i]], ordf(sraw[i]));
}

__global__ void seg_expsum(const float* __restrict__ sraw,
                           const int* __restrict__ batch,
                           const unsigned* __restrict__ smax,
                           float* __restrict__ z, float* __restrict__ sc, int n) {
  int i = blockIdx.x * blockDim.x + threadIdx.x;
  if (i >= n) return;
  int b = batch[i];
  float e = __expf(sraw[i] - unordf(smax[b]));
  sc[i] = e;
  atomicAdd(&z[b], e);
}

// wave per node: gx[batch] += x * (sc/z)
__global__ __launch_bounds__(256) void seg_weighted_sum(
    const float* __restrict__ x, const float* __restrict__ sc,
    const float* __restrict__ z, const int* __restrict__ batch,
    float* __restrict__ gx, int n) {
  int lane = threadIdx.x & 31;
  int i = (blockIdx.x * blockDim.x + threadIdx.x) >> 5;
  if (i >= n) return;
  int b = batch[i];
  float w = sc[i] / z[b];
  float4 xv = ((const float4*)(x + (size_t)i * D_DIM))[lane];
  float* dst = gx + (size_t)b * D_DIM + lane * 4;
  atomicAdd(dst + 0, xv.x * w);
  atomicAdd(dst + 1, xv.y * w);
  atomicAdd(dst + 2, xv.z * w);
  atomicAdd(dst + 3, xv.w * w);
}

// wave per graph: scores[t] = gout_t[b] . a[:,t] + a_bias[t]; softmax over T=3
__global__ __launch_bounds__(256) void attn_softmax(
    const float* __restrict__ gout, const float* __restrict__ a,
    const float* __restrict__ abias, float* __restrict__ sm, int T) {
  int lane = threadIdx.x & 31;
  int b = (blockIdx.x * blockDim.x + threadIdx.x) >> 5;
  if (b >= NB) return;
  float s[3];
  for (int t = 0; t < 3; ++t) {
    float4 g = ((const float4*)(gout + ((size_t)t * NB + b) * D_DIM))[lane];
    int d0 = lane * 4;
    float p = g.x * a[(d0 + 0) * T + t] + g.y * a[(d0 + 1) * T + t] +
              g.z * a[(d0 + 2) * T + t] + g.w * a[(d0 + 3) * T + t];
    s[t] = wave_sum(p) + abias[t];
  }
  if (lane == 0) {
    float m = fmaxf(s[0], fmaxf(s[1], s[2]));
    float e0 = __expf(s[0] - m), e1 = __expf(s[1] - m), e2 = __expf(s[2] - m);
    float Z = e0 + e1 + e2;
    sm[b * 3 + 0] = e0 / Z;
    sm[b * 3 + 1] = e1 / Z;
    sm[b * 3 + 2] = e2 / Z;
  }
}

// wave per node: out = sum_t outs_t * sm[batch][t]
__global__ __launch_bounds__(256) void final_combine(
    const float* __restrict__ outs, const float* __restrict__ sm,
    const int* __restrict__ batch, float* __restrict__ out, int n) {
  int lane = threadIdx.x & 31;
  int i = (blockIdx.x * blockDim.x + threadIdx.x) >> 5;
  if (i >= n) return;
  int b = batch[i];
  float w0 = sm[b * 3 + 0], w1 = sm[b * 3 + 1], w2 = sm[b * 3 + 2];
  size_t nd = (size_t)n * D_DIM;
  float4 o0 = ((const float4*)(outs + (size_t)i * D_DIM))[lane];
  float4 o1 = ((const float4*)(outs + nd + (size_t)i * D_DIM))[lane];
  float4 o2 = ((const float4*)(outs + 2 * nd + (size_t)i * D_DIM))[lane];
  float4 r;
  r.x = o0.x * w0 + o1.x * w1 + o2.x * w2;
  r.y = o0.y * w0 + o1.y * w1 + o2.y * w2;
  r.z = o0.z * w0 + o1.z * w1 + o2.z * w2;
  r.w = o0.w * w0 + o1.w * w1 + o2.w * w2;
  ((float4*)(out + (size_t)i * D_DIM))[lane] = r;
}

// ---------------------------------------------------------------------------
static inline int cdiv(long long a, long long b) { return (int)((a + b - 1) / b); }

extern "C" void kernel_launch(void* const* d_in, const int* in_sizes, int n_in,
                              void* d_out, int out_size, void* d_ws, size_t ws_size,
                              hipStream_t stream) {
  const float* edge_attr = (const float*)d_in[0];
  const int*   lg_edge   = (const int*)d_in[1];
  const int*   batch     = (const int*)d_in[2];
  const float* W_gcn     = (const float*)d_in[3];
  const float* b_gcn     = (const float*)d_in[4];
  const float* W_rel     = (const float*)d_in[5];
  const float* b_rel     = (const float*)d_in[6];
  const float* W_root    = (const float*)d_in[7];
  const float* W_gout    = (const float*)d_in[8];
  const float* b_gout    = (const float*)d_in[9];
  const float* a_vec     = (const float*)d_in[10];
  const float* a_bias    = (const float*)d_in[11];

  const int E  = in_sizes[2];
  const int LE = in_sizes[1] / 2;
  const int T  = in_sizes[3] / (D_DIM * D_DIM);   // 3
  const size_t ND = (size_t)E * D_DIM;

  const int* rowp = lg_edge;
  const int* colp = lg_edge + LE;

  // workspace carve (16B-aligned regions; all counts are multiples of 4)
  char* ws = (char*)d_ws;
  float*    xw     = (float*)ws;                 ws += ND * sizeof(float);
  float*    agg    = (float*)ws;                 ws += ND * sizeof(float);
  float*    outs   = (float*)ws;                 ws += 3 * ND * sizeof(float);
  float*    dinv   = (float*)ws;                 ws += (size_t)E * sizeof(float);
  unsigned* deg    = (unsigned*)ws;              ws += (size_t)E * sizeof(unsigned);
  float*    s_node = (float*)ws;                 ws += (size_t)E * sizeof(float);
  float*    sraw   = (float*)ws;                 ws += (size_t)E * sizeof(float);
  float*    sc     = (float*)ws;                 ws += (size_t)E * sizeof(float);
  unsigned* smax   = (unsigned*)ws;              ws += (size_t)NB * sizeof(unsigned);
  float*    zseg   = (float*)ws;                 ws += (size_t)NB * sizeof(float);
  float*    gx     = (float*)ws;                 ws += 3 * (size_t)NB * D_DIM * sizeof(float);
  float*    gout   = (float*)ws;                 ws += 3 * (size_t)NB * D_DIM * sizeof(float);
  float*    smw    = (float*)ws;                 ws += (size_t)NB * 3 * sizeof(float);
  (void)ws_size; (void)n_in; (void)out_size;

  // --- degree / dinv precompute (iteration invariant) ---
  zero_u32<<<1024, 256, 0, stream>>>(deg, (size_t)E);
  count_deg<<<cdiv(LE, 256), 256, 0, stream>>>(colp, deg, LE);
  make_dinv<<<cdiv(E, 256), 256, 0, stream>>>(deg, dinv, E);

  const float* x = edge_attr;
  for (int t = 0; t < T && t < 3; ++t) {
    float* x_out  = outs + (size_t)t * ND;
    float* gx_t   = gx + (size_t)t * NB * D_DIM;
    float* gout_t = gout + (size_t)t * NB * D_DIM;

    // GCNConv
    gemm128_wmma<0><<<cdiv(E, TILE_ROWS), 256, 0, stream>>>(
        x, W_gcn + (size_t)t * D_DIM * D_DIM, nullptr, xw, E);
    scale_self<<<cdiv((long long)ND, 256), 256, 0, stream>>>(xw, agg, dinv, ND);
    scatter_edges<<<cdiv((long long)LE * 32, 256), 256, 0, stream>>>(
        xw, agg, rowp, colp, dinv, LE);
    bias_relu<<<cdiv((long long)ND, 256), 256, 0, stream>>>(
        edge_attr, agg, b_gcn + (size_t)t * D_DIM, x_out, ND);
    x = x_out;

    // GraphConv score (scalar-first: dot then scatter)
    node_scores<<<cdiv((long long)E * 32, 256), 256, 0, stream>>>(
        x, W_rel, W_root, b_rel, s_node, sraw, E);
    scatter_scalar<<<cdiv(LE, 256), 256, 0, stream>>>(s_node, sraw, rowp, colp, LE);

    // segment softmax over batch + weighted graph readout
    init_seg<<<cdiv(NB * D_DIM, 256), 256, 0, stream>>>(gx_t, smax, zseg, NB * D_DIM);
    seg_max<<<cdiv(E, 256), 256, 0, stream>>>(sraw, batch, smax, E);
    seg_expsum<<<cdiv(E, 256), 256, 0, stream>>>(sraw, batch, smax, zseg, sc, E);
    seg_weighted_sum<<<cdiv((long long)E * 32, 256), 256, 0, stream>>>(
        x, sc, zseg, batch, gx_t, E);

    // gout_t = tanh(gx_t @ W_gout + b_gout)
    gemm128_wmma<1><<<cdiv(NB, TILE_ROWS), 256, 0, stream>>>(
        gx_t, W_gout, b_gout, gout_t, NB);
  }

  // attention over T and final combine
  attn_softmax<<<cdiv((long long)NB * 32, 256), 256, 0, stream>>>(
      gout, a_vec, a_bias, smw, T);
  final_combine<<<cdiv((long long)E * 32, 256), 256, 0, stream>>>(
      outs, smw, batch, (float*)d_out, E);
}